// ChannelAttentionModule_7713761263652
// MI455X (gfx1250) — compile-verified
//
#include <hip/hip_runtime.h>
#include <math.h>

// Problem constants (from the reference)
#define BB 16
#define CC 256
#define HW 16384          // 128*128
#define ROWS (BB * CC)    // 4096 pooling rows
#define CP  (CC + 4)      // LDS row padding: stride 260 -> lane l16 hits bank (4*l16+k)%64, conflict-free

typedef __attribute__((ext_vector_type(2))) float v2f;
typedef __attribute__((ext_vector_type(4))) float v4f;
typedef __attribute__((ext_vector_type(8))) float v8f;

// ---------------------------------------------------------------------------
// Kernel 1: fused global mean+max pool over spatial dims.
// One block per (b,c) row; 256 threads; 64 floats/thread via 16x B128 NT loads.
// This is the memory-bound part: 256 MB @ 23.3 TB/s ~= 11.5 us floor.
// ---------------------------------------------------------------------------
__global__ __launch_bounds__(256) void cam_pool_kernel(
    const float* __restrict__ x,
    float* __restrict__ avg_out,
    float* __restrict__ max_out) {
  const int row = blockIdx.x;                    // b*C + c
  const v4f* __restrict__ p4 =
      (const v4f*)(x + (size_t)row * HW);        // 4096 float4 per row
  const int tid = threadIdx.x;

  float s = 0.0f;
  float m = -INFINITY;
#pragma unroll
  for (int i = 0; i < 16; ++i) {
    // contiguous across lanes -> fully coalesced 128B/wave segments;
    // non-temporal: single-use streaming data, keep it out of L2.
    v4f v = __builtin_nontemporal_load(&p4[tid + i * 256]);
    s += (v.x + v.y) + (v.z + v.w);
    m = fmaxf(m, fmaxf(fmaxf(v.x, v.y), fmaxf(v.z, v.w)));
  }

  // wave32 butterfly reduction
#pragma unroll
  for (int off = 16; off > 0; off >>= 1) {
    s += __shfl_xor(s, off, 32);
    m = fmaxf(m, __shfl_xor(m, off, 32));
  }

  __shared__ float ss[8];
  __shared__ float sm[8];
  const int wave = tid >> 5;
  if ((tid & 31) == 0) { ss[wave] = s; sm[wave] = m; }
  __syncthreads();
  if (tid == 0) {
    float S = ss[0], M = sm[0];
#pragma unroll
    for (int w = 1; w < 8; ++w) { S += ss[w]; M = fmaxf(M, sm[w]); }
    avg_out[row] = S * (1.0f / (float)HW);
    max_out[row] = M;
  }
}

// ---------------------------------------------------------------------------
// Kernel 2: shared MLP on avg & max pooled vectors + sigmoid, in fp32 WMMA.
//   h   = relu(v @ w1^T)        ([16,256] = [16,256]x[256,256]^T)
//   out = sigmoid(h_a @ w2^T + h_m @ w2^T)
// V_WMMA_F32_16X16X4_F32: M=16 (batch), N=16 tile, K stepped by 4.
// One workgroup, 8 waves; wave w owns N-tiles {2w, 2w+1}.
// Fragment packing per CDNA5 ISA 7.12.2 (32-bit A 16x4 / 32-bit C 16x16).
// ---------------------------------------------------------------------------
__global__ __launch_bounds__(256) void cam_mlp_wmma_kernel(
    const float* __restrict__ avg,   // [16][256]
    const float* __restrict__ mx,    // [16][256]
    const float* __restrict__ w1,    // [256][256] row-major (o, c)
    const float* __restrict__ w2,    // [256][256]
    float* __restrict__ out) {       // [16][256]
  __shared__ float h_avg[BB][CP];
  __shared__ float h_max[BB][CP];

  const int tid  = threadIdx.x;
  const int wave = tid >> 5;      // 0..7
  const int lane = tid & 31;
  const int half = lane >> 4;     // 0: K pair {0,1}; 1: K pair {2,3}
  const int l16  = lane & 15;

  // ---------------- Layer 1: v @ w1^T, ReLU -> LDS ----------------
  for (int t = 0; t < 2; ++t) {
    const int ntile = wave * 2 + t;
    const int ocol  = ntile * 16 + l16;        // output channel this lane feeds in B
    v8f acc_a = {};
    v8f acc_m = {};
    for (int k = 0; k < CC; k += 4) {
      const int ka = k + half * 2;
      // A (16x4, M=batch, K=channel): lane row = l16
      v2f a_a, a_m, b;
      a_a.x = avg[l16 * CC + ka];  a_a.y = avg[l16 * CC + ka + 1];
      a_m.x = mx [l16 * CC + ka];  a_m.y = mx [l16 * CC + ka + 1];
      // B (4x16): B[k][n] = w1[n][k]  (w1 row-major -> 2 consecutive floats)
      b.x = w1[ocol * CC + ka];    b.y = w1[ocol * CC + ka + 1];
      acc_a = __builtin_amdgcn_wmma_f32_16x16x4_f32(
          false, a_a, false, b, (short)0, acc_a, false, false);
      acc_m = __builtin_amdgcn_wmma_f32_16x16x4_f32(
          false, a_m, false, b, (short)0, acc_m, false, false);
    }
    // C/D layout: VGPR r -> M = r + 8*half, N = l16 (within tile). ReLU here.
#pragma unroll
    for (int r = 0; r < 8; ++r) {
      const int mrow = r + 8 * half;
      h_avg[mrow][ntile * 16 + l16] = fmaxf(acc_a[r], 0.0f);
      h_max[mrow][ntile * 16 + l16] = fmaxf(acc_m[r], 0.0f);
    }
  }
  __syncthreads();

  // ---------------- Layer 2: h @ w2^T, add branches, sigmoid ----------------
  for (int t = 0; t < 2; ++t) {
    const int ntile = wave * 2 + t;
    const int ocol  = ntile * 16 + l16;
    v8f acc_a = {};
    v8f acc_m = {};
    for (int k = 0; k < CC; k += 4) {
      const int ka = k + half * 2;
      v2f a_a, a_m, b;
      a_a.x = h_avg[l16][ka];  a_a.y = h_avg[l16][ka + 1];
      a_m.x = h_max[l16][ka];  a_m.y = h_max[l16][ka + 1];
      b.x = w2[ocol * CC + ka];  b.y = w2[ocol * CC + ka + 1];
      acc_a = __builtin_amdgcn_wmma_f32_16x16x4_f32(
          false, a_a, false, b, (short)0, acc_a, false, false);
      acc_m = __builtin_amdgcn_wmma_f32_16x16x4_f32(
          false, a_m, false, b, (short)0, acc_m, false, false);
    }
#pragma unroll
    for (int r = 0; r < 8; ++r) {
      const int mrow = r + 8 * half;
      const float z = acc_a[r] + acc_m[r];
      out[mrow * CC + ntile * 16 + l16] = 1.0f / (1.0f + __expf(-z));
    }
  }
}

// ---------------------------------------------------------------------------
// Launch
// ---------------------------------------------------------------------------
extern "C" void kernel_launch(void* const* d_in, const int* in_sizes, int n_in,
                              void* d_out, int out_size, void* d_ws, size_t ws_size,
                              hipStream_t stream) {
  const float* x  = (const float*)d_in[0];   // [16,256,128,128]
  const float* w1 = (const float*)d_in[1];   // [256,256]
  const float* w2 = (const float*)d_in[2];   // [256,256]
  float* out = (float*)d_out;                // [16,256] (as [16,256,1,1])

  float* ws_avg = (float*)d_ws;              // [4096]
  float* ws_max = ws_avg + ROWS;             // [4096]

  cam_pool_kernel<<<ROWS, 256, 0, stream>>>(x, ws_avg, ws_max);
  cam_mlp_wmma_kernel<<<1, 256, 0, stream>>>(ws_avg, ws_max, w1, w2, out);
}